// AGNNConv_62182536511737
// MI455X (gfx1250) — compile-verified
//
#include <hip/hip_runtime.h>
#include <hip/hip_bf16.h>

typedef float v2f __attribute__((ext_vector_type(2)));
typedef float v8f __attribute__((ext_vector_type(8)));

#define DD 64

// ---------------------------------------------------------------------------
// Kernel 1: Xp = X @ W  via V_WMMA_F32_16X16X4_F32.
// One wave computes one 16x16 tile of Xp. W staged in LDS (16 KB).
// ---------------------------------------------------------------------------
__global__ void gemm_xw_wmma(const float* __restrict__ X,
                             const float* __restrict__ W,
                             float* __restrict__ Xp,
                             int N, int rowTiles)
{
    __shared__ float Wl[DD * DD];
    for (int i = threadIdx.x; i < DD * DD; i += blockDim.x) Wl[i] = W[i];
    __syncthreads();

    const int wave = blockIdx.x * (blockDim.x >> 5) + (threadIdx.x >> 5);
    const int lane = threadIdx.x & 31;
    const int tiles = rowTiles * 4;            // 4 column tiles of 16 (D=64)
    if (wave >= tiles) return;                 // wave-uniform exit

    const int rt = wave >> 2, ct = wave & 3;
    const int rBase = rt * 16, cBase = ct * 16;
    const int mm   = lane & 15;                // M (for A) / N (for B,C)
    const int hi   = lane >> 4;
    const int kOff = hi * 2;                   // K sub-offset per half-wave

    int r = rBase + mm;                        // A-row this lane loads
    if (r >= N) r = N - 1;                     // clamp (keeps EXEC full)

    v8f c = {0.f, 0.f, 0.f, 0.f, 0.f, 0.f, 0.f, 0.f};

#pragma unroll
    for (int k = 0; k < DD; k += 4) {
        // A fragment: lane holds X[r, k+kOff .. k+kOff+1]  (8B aligned)
        v2f a = *(const v2f*)(X + (size_t)r * DD + k + kOff);
        // B fragment from LDS: lane holds W[k+kOff, cBase+mm], W[k+kOff+1, cBase+mm]
        v2f b;
        b.x = Wl[(k + kOff)     * DD + cBase + mm];
        b.y = Wl[(k + kOff + 1) * DD + cBase + mm];
        c = __builtin_amdgcn_wmma_f32_16x16x4_f32(
                /*neg_a=*/false, a, /*neg_b=*/false, b,
                /*c_mod=*/(short)0, c, /*reuse_a=*/false, /*reuse_b=*/false);
    }

#pragma unroll
    for (int v = 0; v < 8; ++v) {
        int m = rBase + v + (hi << 3);         // VGPR v -> M = v (+8 for hi half)
        if (m < N) Xp[(size_t)m * DD + cBase + mm] = c[v];
    }
}

// ---------------------------------------------------------------------------
// Kernel 2: fused SDDMM + exp + SpMM. One wave per 8 edges; lane owns 2 features.
// Row-sorted edges: accumulate same-row runs in registers, flush with atomics
// only on row change (avg degree ~12 -> big atomic-traffic reduction).
// ---------------------------------------------------------------------------
__global__ void edge_fused(const float* __restrict__ Xp,
                           const int* __restrict__ row,
                           const int* __restrict__ col,
                           const float* __restrict__ aw_p,
                           float* __restrict__ hout,   // [N,64] pre-zeroed
                           float* __restrict__ rsum,   // [N]    pre-zeroed
                           int E)
{
    const int EPW = 8;                                   // edges per wave
    const int wave = blockIdx.x * (blockDim.x >> 5) + (threadIdx.x >> 5);
    const int lane = threadIdx.x & 31;
    const int base = wave * EPW;
    if (base >= E) return;                               // wave-uniform

    const float aw = aw_p[0];
    const int d0 = lane * 2;

    float acc0 = 0.f, acc1 = 0.f, accS = 0.f;
    int curRow = -1;

    for (int i = 0; i < EPW; ++i) {
        int e = base + i;
        if (e >= E) break;                               // wave-uniform
        int r = row[e], cc = col[e];                     // uniform across wave

        v2f xr = *(const v2f*)(Xp + (size_t)r  * DD + d0);
        v2f xc = *(const v2f*)(Xp + (size_t)cc * DD + d0);

        float part = xr.x * xc.x + xr.y * xc.y;
        // full wave32 reduction
        for (int m = 16; m >= 1; m >>= 1)
            part += __shfl_xor(part, m, 32);

        float att = __expf(aw * part);

        if (r != curRow) {                               // uniform branch
            if (curRow >= 0) {
                atomicAdd(&hout[(size_t)curRow * DD + d0],     acc0);
                atomicAdd(&hout[(size_t)curRow * DD + d0 + 1], acc1);
                if (lane == 0) atomicAdd(&rsum[curRow], accS);
            }
            curRow = r; acc0 = acc1 = accS = 0.f;
        }
        acc0 += att * xc.x;
        acc1 += att * xc.y;
        accS += att;
    }
    if (curRow >= 0) {
        atomicAdd(&hout[(size_t)curRow * DD + d0],     acc0);
        atomicAdd(&hout[(size_t)curRow * DD + d0 + 1], acc1);
        if (lane == 0) atomicAdd(&rsum[curRow], accS);
    }
}

// ---------------------------------------------------------------------------
// Kernel 3: out[n,d] /= rows_sum[n]
// ---------------------------------------------------------------------------
__global__ void scale_rows(float* __restrict__ hout,
                           const float* __restrict__ rsum,
                           int total)
{
    int i = blockIdx.x * blockDim.x + threadIdx.x;
    if (i < total) hout[i] /= rsum[i >> 6];
}

// ---------------------------------------------------------------------------
extern "C" void kernel_launch(void* const* d_in, const int* in_sizes, int n_in,
                              void* d_out, int out_size, void* d_ws, size_t ws_size,
                              hipStream_t stream)
{
    const float* X   = (const float*)d_in[0];
    const float* W   = (const float*)d_in[1];
    const float* aw  = (const float*)d_in[2];
    const int*   row = (const int*)d_in[3];
    const int*   col = (const int*)d_in[4];
    float* out = (float*)d_out;

    const int N = in_sizes[0] / DD;
    const int E = in_sizes[3];

    // workspace layout: Xp [N*64 floats] | rows_sum [N floats]
    float* Xp   = (float*)d_ws;
    float* rsum = Xp + (size_t)N * DD;

    hipMemsetAsync(out,  0, (size_t)N * DD * sizeof(float), stream);
    hipMemsetAsync(rsum, 0, (size_t)N * sizeof(float), stream);

    // GEMM: (N/16)*4 tiles, 8 waves (= 8 tiles) per 256-thread block
    const int rowTiles = (N + 15) / 16;
    const int tiles    = rowTiles * 4;
    const int gBlocks  = (tiles + 7) / 8;
    gemm_xw_wmma<<<gBlocks, 256, 0, stream>>>(X, W, Xp, N, rowTiles);

    // Edge phase: 1 wave per 8 edges, 8 waves per block
    const int waves   = (E + 7) / 8;
    const int eBlocks = (waves + 7) / 8;
    edge_fused<<<eBlocks, 256, 0, stream>>>(Xp, row, col, aw, out, rsum, E);

    // Normalize
    const int total = N * DD;
    scale_rows<<<(total + 255) / 256, 256, 0, stream>>>(out, rsum, total);
}